// RNNTLossWrapper_62775241999019
// MI455X (gfx1250) — compile-verified
//
#include <hip/hip_runtime.h>
#include <hip/hip_bf16.h>

// RNNT loss, MI455X (gfx1250).
// Phase 1 (bandwidth-bound, ~541 MB stream): per-row max/logsumexp over V=256,
//   emitting only blank/emit log-probs. Streamed via async global->LDS copies
//   (double-buffered per wave) using the gfx1250 async-to-LDS engine.
// Phase 2: anti-diagonal wavefront DP over the (T x U+1) lattice, 1 block/batch.
// Phase 3: mean reduction.

#define T_  256
#define B_  16
#define U1_ 129
#define U_  128
#define V_  256
#define BLANK_ 255

#define NROWS   (T_ * B_ * U1_)      // 528384
#define TILE_ROWS 4
#define NTILES  (NROWS / TILE_ROWS)  // 132096

#ifndef __has_builtin
#define __has_builtin(x) 0
#endif

#if __has_builtin(__builtin_amdgcn_global_load_async_to_lds_b128)
#define HAVE_ASYNC_LDS 1
#else
#define HAVE_ASYNC_LDS 0
#endif

#if __has_builtin(__builtin_amdgcn_s_wait_asynccnt)
#define WAIT_ASYNC(n) __builtin_amdgcn_s_wait_asynccnt(n)
#else
#define WAIT_ASYNC(n) asm volatile("s_wait_asynccnt %0" :: "i"(n) : "memory")
#endif

#define AS1 __attribute__((address_space(1)))
#define AS3 __attribute__((address_space(3)))

typedef int v4i __attribute__((ext_vector_type(4)));

__device__ __forceinline__ float pick4(const float4& v, int c) {
  return (c & 1) ? ((c & 2) ? v.w : v.y) : ((c & 2) ? v.z : v.x);
}

// Reduce one row (256 floats held as two float4 per lane) to logsumexp,
// write blank/emit outputs. All 32 lanes participate.
__device__ __forceinline__ void row_reduce_emit(
    int row, float4 v0, float4 v1, int lane,
    const int* __restrict__ text,
    float* __restrict__ blankO, float* __restrict__ emitO) {
  // decode row -> (t, b, u)
  int t   = row / (B_ * U1_);
  int rem = row - t * (B_ * U1_);
  int b   = rem / U1_;
  int u   = rem - b * U1_;

  float m = fmaxf(fmaxf(fmaxf(v0.x, v0.y), fmaxf(v0.z, v0.w)),
                  fmaxf(fmaxf(v1.x, v1.y), fmaxf(v1.z, v1.w)));
  #pragma unroll
  for (int o = 16; o >= 1; o >>= 1) m = fmaxf(m, __shfl_xor(m, o, 32));

  float s = __expf(v0.x - m) + __expf(v0.y - m) + __expf(v0.z - m) + __expf(v0.w - m)
          + __expf(v1.x - m) + __expf(v1.y - m) + __expf(v1.z - m) + __expf(v1.w - m);
  #pragma unroll
  for (int o = 16; o >= 1; o >>= 1) s += __shfl_xor(s, o, 32);

  float lse = m + __logf(s);

  // blank = element 255 -> lane 31, second chunk, component 3
  if (lane == 31) blankO[row] = v1.w - lse;

  // emit = element text[b][u], only for u < U
  if (u < U_) {
    int k = text[b * U_ + u];            // uniform across the wave
    int kc = k & 127;
    if (lane == (kc >> 2)) {
      float xv = (k >= 128) ? pick4(v1, k & 3) : pick4(v0, k & 3);
      emitO[(t * B_ + b) * U_ + u] = xv - lse;
    }
  }
}

__global__ void __launch_bounds__(256)
k_softmax_stats(const float* __restrict__ lp, const int* __restrict__ text,
                float* __restrict__ blankO, float* __restrict__ emitO) {
  const int lane = threadIdx.x & 31;
  const int warp = threadIdx.x >> 5;
  const int nw   = gridDim.x * (blockDim.x >> 5);
  const int wid  = blockIdx.x * (blockDim.x >> 5) + warp;

#if HAVE_ASYNC_LDS
  // per-wave double buffer: 2 x 4KB (one 4-row tile each)
  __shared__ __align__(16) char smem[8 * 2 * 4096];
  char* myLds = smem + warp * 8192;

  const char* gbase = (const char*)lp;

  auto issue = [&](int tile, int buf) {
    const char* g = gbase + (size_t)tile * (TILE_ROWS * V_ * 4);
    char* l = myLds + buf * 4096;
    #pragma unroll
    for (int i = 0; i < 8; ++i) {
      __builtin_amdgcn_global_load_async_to_lds_b128(
          (AS1 v4i*)(g + i * 512 + lane * 16),
          (AS3 v4i*)(l + i * 512 + lane * 16), 0, 0);
    }
  };

  auto compute = [&](int tile, int buf) {
    const char* l = myLds + buf * 4096;
    #pragma unroll
    for (int j = 0; j < TILE_ROWS; ++j) {
      const char* rl = l + j * 1024;
      float4 v0 = *(const float4*)(rl + lane * 16);
      float4 v1 = *(const float4*)(rl + 512 + lane * 16);
      row_reduce_emit(tile * TILE_ROWS + j, v0, v1, lane, text, blankO, emitO);
    }
  };

  int tile = wid;
  if (tile < NTILES) issue(tile, 0);
  int prevTile = tile, pbuf = 0;
  tile += nw;
  while (prevTile < NTILES) {
    bool more = (tile < NTILES);
    if (more) issue(tile, pbuf ^ 1);
    if (more) { WAIT_ASYNC(8); } else { WAIT_ASYNC(0); }
    compute(prevTile, pbuf);
    prevTile = tile;
    pbuf ^= 1;
    tile += nw;
  }
#else
  // fallback: direct 128-bit global loads, grid-stride over rows
  for (int row = wid; row < NROWS; row += nw) {
    const float4* rp = (const float4*)(lp + (size_t)row * V_);
    float4 v0 = rp[lane];
    float4 v1 = rp[32 + lane];
    row_reduce_emit(row, v0, v1, lane, text, blankO, emitO);
  }
#endif
}

__device__ __forceinline__ float lae(float a, float b) {
  float mx = fmaxf(a, b), mn = fminf(a, b);
  return mx + __logf(1.0f + __expf(mn - mx));
}

// One block per batch. Anti-diagonal wavefront over the (T x U+1) lattice.
__global__ void __launch_bounds__(160)
k_dp(const float* __restrict__ blank, const float* __restrict__ emit,
     const int* __restrict__ lplen, const int* __restrict__ tlen,
     float* __restrict__ loss) {
  const int b = blockIdx.x;
  const int u = threadIdx.x;      // 0..159; active when u <= U_
  const int tb = lplen[b] - 1;
  const int ub = tlen[b];

  __shared__ float diag[2][U1_ + 3];

  float ll = 0.0f;
  bool have = false;

  for (int d = 0; d < T_ + U_; ++d) {
    float* cur  = diag[d & 1];
    float* prev = diag[(d & 1) ^ 1];
    int t = d - u;
    bool valid = (u <= U_) && (t >= 0) && (t < T_);
    float r = 0.0f;
    if (valid) {
      if (d == 0) {
        r = 0.0f;                                         // alpha[0][0]
      } else if (t == 0) {
        r = prev[u - 1] + emit[(0 * B_ + b) * U_ + (u - 1)];        // cumsum row
      } else if (u == 0) {
        r = prev[0] + blank[((t - 1) * B_ + b) * U1_ + 0];
      } else {
        float va = prev[u]     + blank[((t - 1) * B_ + b) * U1_ + u];
        float vb = prev[u - 1] + emit[(t * B_ + b) * U_ + (u - 1)];
        r = lae(va, vb);
      }
      cur[u] = r;
      if (t == tb && u == ub) { ll = r + blank[(tb * B_ + b) * U1_ + ub]; have = true; }
    }
    __syncthreads();
  }

  if (have) loss[b] = ll;
}

__global__ void k_final(const float* __restrict__ loss, float* __restrict__ out) {
  if (threadIdx.x == 0) {
    float s = 0.0f;
    #pragma unroll
    for (int b = 0; b < B_; ++b) s += loss[b];
    out[0] = -s / (float)B_;
  }
}

extern "C" void kernel_launch(void* const* d_in, const int* in_sizes, int n_in,
                              void* d_out, int out_size, void* d_ws, size_t ws_size,
                              hipStream_t stream) {
  const float* lp   = (const float*)d_in[0];
  const int*  lplen = (const int*)d_in[1];
  const int*  text  = (const int*)d_in[2];
  const int*  tlen  = (const int*)d_in[3];

  float* blankA = (float*)d_ws;                       // T*B*U1 floats
  float* emitA  = blankA + (size_t)T_ * B_ * U1_;     // T*B*U  floats
  float* lossA  = emitA + (size_t)T_ * B_ * U_;       // B floats

  k_softmax_stats<<<1024, 256, 0, stream>>>(lp, text, blankA, emitA);
  k_dp<<<B_, 160, 0, stream>>>(blankA, emitA, lplen, tlen, lossA);
  k_final<<<1, 32, 0, stream>>>(lossA, (float*)d_out);
}